// _System_general_8581344657603
// MI455X (gfx1250) — compile-verified
//
#include <hip/hip_runtime.h>
#include <math.h>

// MI455X / gfx1250 fully-fused kernel for the tanh forward-substitution SSM:
//   V  = xi@C1^T + u@D12^T + bv
//   w  = blocked triangular tanh recurrence over strict-lower D11
//   out= xi@A^T + w@B1^T + u@B2^T + bx
// One workgroup = 64 batch rows; xi-tile and V/w-tile live in LDS (CDNA5 has
// 320KB/WGP), so the 128MB of v0/w intermediates never touch HBM.
// All dense math uses V_WMMA_F32_16X16X4_F32 (exact fp32, matches reference).

typedef __attribute__((ext_vector_type(2))) float v2f;
typedef __attribute__((ext_vector_type(4))) float v4f;
typedef __attribute__((ext_vector_type(8))) float v8f;

constexpr int BATCH = 32768;
constexpr int NX = 512;
constexpr int NU = 256;
constexpr int NQ = 512;
constexpr int BT = 64;     // batch rows per workgroup
constexpr int LD = 516;    // padded LDS row stride (floats) -> conflict-free frag loads

union V8 { v8f v; float f[8]; };

__device__ __forceinline__ v8f wmma4(v2f a, v2f b, v8f c) {
  // D = A(16x4,f32) x B(4x16,f32) + C(16x16,f32)
  return __builtin_amdgcn_wmma_f32_16x16x4_f32(
      /*neg_a=*/false, a, /*neg_b=*/false, b,
      /*c_mod=*/(short)0, c, /*reuse_a=*/false, /*reuse_b=*/false);
}

__global__ __launch_bounds__(256)
void ssm_fused_kernel(const float* __restrict__ xi, const float* __restrict__ u,
                      const float* __restrict__ Aw, const float* __restrict__ B1,
                      const float* __restrict__ B2, const float* __restrict__ C1,
                      const float* __restrict__ D11, const float* __restrict__ D12,
                      const float* __restrict__ bx, const float* __restrict__ bv,
                      float* __restrict__ out)
{
  __shared__ float Xs[BT * LD];      // xi tile      (64 x 512)
  __shared__ float Vw[BT * LD];      // V, then w in place
  __shared__ float Ps[4 * BT * 16];  // K-split partials for off-diagonal GEMM

  const int tid  = threadIdx.x;
  const int lane = tid & 31;
  const int wv   = tid >> 5;          // wave 0..7
  const int m    = lane & 15;         // M (or N) index within a 16-wide frag
  const int kh   = lane >> 4;         // K-half selector for A/B frags
  const int row0 = blockIdx.x * BT;   // first batch row of this tile

  // ---------------- stage xi tile into LDS (float4 coalesced) ----------------
  for (int idx = tid; idx < BT * (NX / 4); idx += 256) {
    const int b  = idx >> 7;              // NX/4 = 128 float4 per row
    const int kc = (idx & 127) << 2;
    v4f val = *(const v4f*)(xi + (size_t)(row0 + b) * NX + kc);
    *(v4f*)&Xs[b * LD + kc] = val;
  }
  __syncthreads();

  // A-frag row pointers for the 4 M-tiles (rows m0 + m)
  const float* xa[4];
  const float* wa[4];
  const float* ua[4];
  #pragma unroll
  for (int mt = 0; mt < 4; ++mt) {
    xa[mt] = &Xs[(mt * 16 + m) * LD];
    wa[mt] = &Vw[(mt * 16 + m) * LD];
    ua[mt] = u + (size_t)(row0 + mt * 16 + m) * NU;
  }

  // ---------------- phase 1: V = xi@C1^T + u@D12^T + bv ----------------
  // Each wave owns 4 of the 32 column tiles, register-blocks all 4 M-tiles so
  // every 8B weight load feeds 4 WMMAs.
  for (int t = 0; t < 4; ++t) {
    const int nc = (wv * 4 + t) * 16 + m;          // this lane's output column
    V8 acc[4] = {};
    const float* brow = C1 + (size_t)nc * NX;
    for (int k = 0; k < NX; k += 4) {
      v2f b = *(const v2f*)(brow + k + 2 * kh);
      #pragma unroll
      for (int mt = 0; mt < 4; ++mt) {
        v2f a = *(const v2f*)(xa[mt] + k + 2 * kh);
        acc[mt].v = wmma4(a, b, acc[mt].v);
      }
    }
    const float* brow2 = D12 + (size_t)nc * NU;
    for (int k = 0; k < NU; k += 4) {
      v2f b = *(const v2f*)(brow2 + k + 2 * kh);
      #pragma unroll
      for (int mt = 0; mt < 4; ++mt) {
        v2f a = *(const v2f*)(ua[mt] + k + 2 * kh);
        acc[mt].v = wmma4(a, b, acc[mt].v);
      }
    }
    const float bvv = bv[nc];
    #pragma unroll
    for (int mt = 0; mt < 4; ++mt) {
      const int rbase = mt * 16 + kh * 8;          // C/D layout: vgpr r -> rows r, r+8
      #pragma unroll
      for (int r = 0; r < 8; ++r)
        Vw[(rbase + r) * LD + nc] = acc[mt].f[r] + bvv;
    }
  }

  // ---------------- phase 2: blocked tanh forward substitution ----------------
  // w[:,q] = tanh(V[:,q] + sum_{j<q} D11[q,j] * w[:,j]); 16-column blocks.
  for (int kb = 0; kb < 32; ++kb) {
    __syncthreads();                                // prev block's w visible
    const int q0 = kb * 16;
    {
      // off-diagonal: Ps[kq] = w[:, kbeg:kend] @ L_block^T, K split 4 ways
      const int mp = wv >> 2;                       // M-tile pair 0..1
      const int kq = wv & 3;                        // K quarter  0..3
      const int kbeg = kq * 4 * kb;
      const int kend = kbeg + 4 * kb;               // 4*kb elems, multiple of 4
      V8 acc[2] = {};
      const float* lrow = D11 + (size_t)(q0 + m) * NQ;
      for (int k = kbeg; k < kend; k += 4) {
        v2f b = *(const v2f*)(lrow + k + 2 * kh);
        #pragma unroll
        for (int mt = 0; mt < 2; ++mt) {
          v2f a = *(const v2f*)(wa[mp * 2 + mt] + k + 2 * kh);
          acc[mt].v = wmma4(a, b, acc[mt].v);
        }
      }
      float* ps = &Ps[kq * BT * 16];
      #pragma unroll
      for (int mt = 0; mt < 2; ++mt) {
        const int rbase = (mp * 2 + mt) * 16 + kh * 8;
        #pragma unroll
        for (int r = 0; r < 8; ++r)
          ps[(rbase + r) * 16 + m] = acc[mt].f[r];  // kb==0 -> zeros (empty K)
      }
    }
    __syncthreads();
    // diagonal 16x16 block: serial in q, parallel over 64 batch rows
    if (tid < BT) {
      float wl[16];
      float* vrow = &Vw[tid * LD + q0];
      const float* p0 = &Ps[0 * BT * 16 + tid * 16];
      const float* p1 = &Ps[1 * BT * 16 + tid * 16];
      const float* p2 = &Ps[2 * BT * 16 + tid * 16];
      const float* p3 = &Ps[3 * BT * 16 + tid * 16];
      for (int qq = 0; qq < 16; ++qq) {
        float val = vrow[qq] + p0[qq] + p1[qq] + p2[qq] + p3[qq];
        const float* drow = D11 + (size_t)(q0 + qq) * NQ + q0;  // strict lower
        for (int j = 0; j < qq; ++j)
          val += drow[j] * wl[j];
        wl[qq] = tanhf(val);
        vrow[qq] = wl[qq];
      }
    }
  }
  __syncthreads();

  // ---------------- phase 3: out = xi@A^T + w@B1^T + u@B2^T + bx ----------------
  for (int t = 0; t < 4; ++t) {
    const int nc = (wv * 4 + t) * 16 + m;
    V8 acc[4] = {};
    const float* awr = Aw + (size_t)nc * NX;
    const float* b1r = B1 + (size_t)nc * NQ;
    const float* b2r = B2 + (size_t)nc * NU;
    for (int k = 0; k < NX; k += 4) {
      v2f b = *(const v2f*)(awr + k + 2 * kh);
      #pragma unroll
      for (int mt = 0; mt < 4; ++mt) {
        v2f a = *(const v2f*)(xa[mt] + k + 2 * kh);
        acc[mt].v = wmma4(a, b, acc[mt].v);
      }
    }
    for (int k = 0; k < NQ; k += 4) {
      v2f b = *(const v2f*)(b1r + k + 2 * kh);
      #pragma unroll
      for (int mt = 0; mt < 4; ++mt) {
        v2f a = *(const v2f*)(wa[mt] + k + 2 * kh);
        acc[mt].v = wmma4(a, b, acc[mt].v);
      }
    }
    for (int k = 0; k < NU; k += 4) {
      v2f b = *(const v2f*)(b2r + k + 2 * kh);
      #pragma unroll
      for (int mt = 0; mt < 4; ++mt) {
        v2f a = *(const v2f*)(ua[mt] + k + 2 * kh);
        acc[mt].v = wmma4(a, b, acc[mt].v);
      }
    }
    const float bxv = bx[nc];
    #pragma unroll
    for (int mt = 0; mt < 4; ++mt) {
      const int rbase = mt * 16 + kh * 8;
      #pragma unroll
      for (int r = 0; r < 8; ++r)
        out[(size_t)(row0 + rbase + r) * NX + nc] = acc[mt].f[r] + bxv;
    }
  }
}

extern "C" void kernel_launch(void* const* d_in, const int* in_sizes, int n_in,
                              void* d_out, int out_size, void* d_ws, size_t ws_size,
                              hipStream_t stream) {
  // setup_inputs order: t, xi, u, A, B1, B2, C1, D11, D12, bx, bv
  const float* xi  = (const float*)d_in[1];
  const float* u   = (const float*)d_in[2];
  const float* Aw  = (const float*)d_in[3];
  const float* B1  = (const float*)d_in[4];
  const float* B2  = (const float*)d_in[5];
  const float* C1  = (const float*)d_in[6];
  const float* D11 = (const float*)d_in[7];
  const float* D12 = (const float*)d_in[8];
  const float* bx  = (const float*)d_in[9];
  const float* bv  = (const float*)d_in[10];
  float* outp = (float*)d_out;

  dim3 grid(BATCH / BT);   // 512 batch tiles
  dim3 block(256);         // 8 wave32
  ssm_fused_kernel<<<grid, block, 0, stream>>>(xi, u, Aw, B1, B2, C1, D11, D12,
                                               bx, bv, outp);
}